// GCN_20023137534525
// MI455X (gfx1250) — compile-verified
//
#include <hip/hip_runtime.h>
#include <hip/hip_bf16.h>
#include <math.h>

#define N_NODES 100000
#define N_EDGES 3200000
#define F_IN    128
#define HID     8
#define N_CLS   4
#define NTILES  (N_NODES / 16)   // 6250 exact

typedef float v2f __attribute__((ext_vector_type(2)));
typedef float v8f __attribute__((ext_vector_type(8)));

__device__ __forceinline__ void atomic_add_f32(float* p, float v) {
    __hip_atomic_fetch_add(p, v, __ATOMIC_RELAXED, __HIP_MEMORY_SCOPE_AGENT);
}

// ---------------- degree / dinv ----------------
__global__ void k_zero_cnt(int* cnt, int n) {
    int i = blockIdx.x * blockDim.x + threadIdx.x;
    if (i < n) cnt[i] = 0;
}

__global__ void k_count(const int* __restrict__ dst, int* cnt, int e) {
    for (int i = blockIdx.x * blockDim.x + threadIdx.x; i < e;
         i += gridDim.x * blockDim.x)
        atomicAdd(&cnt[dst[i]], 1);
}

__global__ void k_dinv(const int* __restrict__ cnt, float* dinv, int n) {
    int i = blockIdx.x * blockDim.x + threadIdx.x;
    if (i < n) dinv[i] = rsqrtf((float)cnt[i] + 1.0f);   // +1 self-loop; deg>=1
}

// ---------------- layer-1 GEMM via f32 WMMA: hs = (x @ W1) * dinv[row] -----
// One wave per 16-row tile. A: 16x4 f32 (lanes 0-15: M, K={kb,kb+1};
// lanes 16-31: M, K={kb+2,kb+3}). B: 4x16 (lane&15 = N, 2 VGPRs = K pair).
// W staged in LDS transposed + zero-padded to 16 columns so B-fragment loads
// are unconditional contiguous b64 reads (no per-step exec divergence).
__global__ void __launch_bounds__(256)
k_gemm1_wmma(const float* __restrict__ x, const float* __restrict__ W1,
             const float* __restrict__ dinv, float* __restrict__ hs) {
    __shared__ float Wt[16 * F_IN];                      // [n][k], 8 KB
    for (int i = threadIdx.x; i < 16 * F_IN; i += blockDim.x) {
        int n = i >> 7;                                  // i / F_IN
        int k = i & (F_IN - 1);
        Wt[i] = (n < HID) ? W1[k * HID + n] : 0.0f;      // pad N=8..15 with 0
    }
    __syncthreads();

    int wave = threadIdx.x >> 5;
    int lane = threadIdx.x & 31;
    int tile = blockIdx.x * 8 + wave;
    if (tile >= NTILES) return;                          // wave-uniform exit

    int half = lane >> 4;            // 0: K pair {0,1}, 1: K pair {2,3}
    int l16  = lane & 15;
    int koff = half * 2;

    // Preload all 32 B fragments from LDS into registers (contiguous b64s).
    v2f bfrag[F_IN / 4];
    const float* wrow = Wt + l16 * F_IN;
#pragma unroll
    for (int s = 0; s < F_IN / 4; ++s) {
        bfrag[s].x = wrow[s * 4 + koff];
        bfrag[s].y = wrow[s * 4 + koff + 1];
    }

    // Hot loop: pure global_load_b64 (A) + chained v_wmma.
    const float* xrow = x + (size_t)(tile * 16 + l16) * F_IN;
    v8f c = {};
#pragma unroll
    for (int s = 0; s < F_IN / 4; ++s) {
        v2f a;
        a.x = xrow[s * 4 + koff];
        a.y = xrow[s * 4 + koff + 1];
        c = __builtin_amdgcn_wmma_f32_16x16x4_f32(
                false, a, false, bfrag[s], (short)0, c, false, false);
    }

    if (l16 < HID) {                 // padded columns 8..15 must not store
#pragma unroll
        for (int r = 0; r < 8; ++r) {
            int row = tile * 16 + r + half * 8;
            hs[(size_t)row * HID + l16] = c[r] * dinv[row];
        }
    }
}

// ---------------- acc = hs (self-loop contribution) ----------------
__global__ void k_copy4(float4* __restrict__ dst, const float4* __restrict__ src,
                        int n4) {
    for (int i = blockIdx.x * blockDim.x + threadIdx.x; i < n4;
         i += gridDim.x * blockDim.x)
        dst[i] = src[i];
}

// ---------------- edge scatter: acc[dst] += hs[src] ----------------
__global__ void k_scatter(const int* __restrict__ src, const int* __restrict__ dst,
                          const float* __restrict__ hs, float* acc, int e) {
    long long total  = (long long)e * HID;
    long long stride = (long long)gridDim.x * blockDim.x;
    for (long long i = (long long)blockIdx.x * blockDim.x + threadIdx.x;
         i < total; i += stride) {
        int ei = (int)(i >> 3);
        int f  = (int)(i & 7);
        int s = src[ei], d = dst[ei];
        atomic_add_f32(&acc[(size_t)d * HID + f], hs[(size_t)s * HID + f]);
    }
}

// ---- hs = (tanh(acc*dinv + b) @ Wn) * dinv  (8x8 GEMM fused, W in LDS) ----
__global__ void __launch_bounds__(256)
k_fin8(const float* __restrict__ acc, const float* __restrict__ dinv,
       const float* __restrict__ b, const float* __restrict__ Wn,
       float* __restrict__ hs, int n) {
    __shared__ float Wl[HID * HID];
    __shared__ float bl[HID];
    if (threadIdx.x < HID * HID) Wl[threadIdx.x] = Wn[threadIdx.x];
    if (threadIdx.x < HID)       bl[threadIdx.x] = b[threadIdx.x];
    __syncthreads();
    int i = blockIdx.x * blockDim.x + threadIdx.x;
    if (i >= n) return;
    float di = dinv[i];
    float v[HID];
#pragma unroll
    for (int f = 0; f < HID; ++f)
        v[f] = tanhf(acc[(size_t)i * HID + f] * di + bl[f]);
#pragma unroll
    for (int j = 0; j < HID; ++j) {
        float s = 0.0f;
#pragma unroll
        for (int f = 0; f < HID; ++f) s += v[f] * Wl[f * HID + j];
        hs[(size_t)i * HID + j] = s * di;
    }
}

// ---- out = tanh(acc*dinv + b3) @ Wc + bc ----
__global__ void __launch_bounds__(256)
k_fincls(const float* __restrict__ acc, const float* __restrict__ dinv,
         const float* __restrict__ b3, const float* __restrict__ Wc,
         const float* __restrict__ bc, float* __restrict__ out, int n) {
    __shared__ float Wl[HID * N_CLS];
    __shared__ float bl[HID];
    __shared__ float bcl[N_CLS];
    if (threadIdx.x < HID * N_CLS) Wl[threadIdx.x]  = Wc[threadIdx.x];
    if (threadIdx.x < HID)         bl[threadIdx.x]  = b3[threadIdx.x];
    if (threadIdx.x < N_CLS)       bcl[threadIdx.x] = bc[threadIdx.x];
    __syncthreads();
    int i = blockIdx.x * blockDim.x + threadIdx.x;
    if (i >= n) return;
    float di = dinv[i];
    float v[HID];
#pragma unroll
    for (int f = 0; f < HID; ++f)
        v[f] = tanhf(acc[(size_t)i * HID + f] * di + bl[f]);
#pragma unroll
    for (int j = 0; j < N_CLS; ++j) {
        float s = bcl[j];
#pragma unroll
        for (int f = 0; f < HID; ++f) s += v[f] * Wl[f * N_CLS + j];
        out[(size_t)i * N_CLS + j] = s;
    }
}

extern "C" void kernel_launch(void* const* d_in, const int* in_sizes, int n_in,
                              void* d_out, int out_size, void* d_ws, size_t ws_size,
                              hipStream_t stream) {
    const float* x   = (const float*)d_in[0];
    const int*   eix = (const int*)d_in[1];     // int32 per harness convention
    const float* W1  = (const float*)d_in[2];
    const float* b1  = (const float*)d_in[3];
    const float* W2  = (const float*)d_in[4];
    const float* b2  = (const float*)d_in[5];
    const float* W3  = (const float*)d_in[6];
    const float* b3  = (const float*)d_in[7];
    const float* Wc  = (const float*)d_in[8];
    const float* bc  = (const float*)d_in[9];
    float* out = (float*)d_out;

    const int* esrc = eix;
    const int* edst = eix + N_EDGES;

    char* ws = (char*)d_ws;
    float* dinv = (float*)ws;  ws += (size_t)N_NODES * sizeof(float);
    int*   cnt  = (int*)ws;    ws += (size_t)N_NODES * sizeof(int);
    float* hs   = (float*)ws;  ws += (size_t)N_NODES * HID * sizeof(float);
    float* acc  = (float*)ws;

    const dim3 B(256);
    const int nb_n  = (N_NODES + 255) / 256;
    const int nb_g  = (NTILES + 7) / 8;
    const int n4    = N_NODES * HID / 4;

    // degree + dinv
    k_zero_cnt<<<nb_n, B, 0, stream>>>(cnt, N_NODES);
    k_count<<<2048, B, 0, stream>>>(edst, cnt, N_EDGES);
    k_dinv<<<nb_n, B, 0, stream>>>(cnt, dinv, N_NODES);

    // layer 1: hs = (x@W1)*dinv via WMMA; acc = hs (self-loops); scatter
    k_gemm1_wmma<<<nb_g, B, 0, stream>>>(x, W1, dinv, hs);
    k_copy4<<<800, B, 0, stream>>>((float4*)acc, (const float4*)hs, n4);
    k_scatter<<<8192, B, 0, stream>>>(esrc, edst, hs, acc, N_EDGES);

    // layer 2
    k_fin8<<<nb_n, B, 0, stream>>>(acc, dinv, b1, W2, hs, N_NODES);
    k_copy4<<<800, B, 0, stream>>>((float4*)acc, (const float4*)hs, n4);
    k_scatter<<<8192, B, 0, stream>>>(esrc, edst, hs, acc, N_EDGES);

    // layer 3
    k_fin8<<<nb_n, B, 0, stream>>>(acc, dinv, b2, W3, hs, N_NODES);
    k_copy4<<<800, B, 0, stream>>>((float4*)acc, (const float4*)hs, n4);
    k_scatter<<<8192, B, 0, stream>>>(esrc, edst, hs, acc, N_EDGES);

    // classifier
    k_fincls<<<nb_n, B, 0, stream>>>(acc, dinv, b3, Wc, bc, out, N_NODES);
}